// SinkhornLayer_13666585936022
// MI455X (gfx1250) — compile-verified
//
#include <hip/hip_runtime.h>
#include <cstdint>

// Sinkhorn (21 iters) over 65536 independent 36x36 matrices, T = 0.01.
// log_alpha stays of the form  s_ij - u_i - v_j  (s = 100*x), so we only
// iterate the two 36-vectors per matrix, all in base-2 log domain:
//   U_i = m + log2( sum_j 2^(K*x_ij - V_j - m) ),  K = 100*log2(e)
//   V_j = m + log2( sum_i 2^(K*x_ij - U_i - m) )
//   out = 2^(K*x_ij - U_i - V_j)
// CDNA5 paths used: global_load_async_to_lds_b32 (ASYNCcnt) for the
// global->LDS staging, s_wait_asynccnt, v_exp_f32/v_log_f32, NT stores.

#define N_ITERS         21
#define MATS_PER_BLOCK  8
#define NTHREADS        288              // 8 matrices * 36 workers = 9 wave32s
#define ROW_STRIDE      37               // 36 + 1 pad dword -> conflict-free banks
#define MAT_STRIDE      (36 * ROW_STRIDE)          // 1332 dwords / matrix
#define ELEMS_PER_MAT   1296
#define ELEMS_PER_BLOCK (MATS_PER_BLOCK * ELEMS_PER_MAT)   // 10368 = 36 * 288

__global__ __launch_bounds__(NTHREADS)
void sinkhorn36_kernel(const float* __restrict__ in, float* __restrict__ out) {
  __shared__ float sx[MATS_PER_BLOCK * MAT_STRIDE];   // 42624 B
  __shared__ float sU[MATS_PER_BLOCK][40];
  __shared__ float sV[MATS_PER_BLOCK][40];

  const int tid  = (int)threadIdx.x;
  const int mat  = tid / 36;
  const int role = tid - mat * 36;

  const float K   = 144.26950408889634f;   // (1/T) * log2(e)
  const float NEG = -3.4e38f;

  const size_t blockBase = (size_t)blockIdx.x * ELEMS_PER_BLOCK;
  const float* gin = in + blockBase;

  const unsigned sxBase = (unsigned)(uintptr_t)(&sx[0]);  // addrspace(3) offset

  // ---- async global -> LDS with padded-row scatter (CDNA5 ASYNCcnt path) ----
  {
    int m = 0, i = mat, j = role;        // f = tid + 288k => j const, i += 8 mod 36
    unsigned voff = (unsigned)(tid * 4);
    #pragma unroll
    for (int k = 0; k < 36; ++k) {
      unsigned lds = sxBase + (unsigned)((m * MAT_STRIDE + i * ROW_STRIDE + j) * 4);
      asm volatile("global_load_async_to_lds_b32 %0, %1, %2"
                   :: "v"(lds), "v"(voff), "s"(gin) : "memory");
      voff += NTHREADS * 4;
      i += 8; if (i >= 36) { i -= 36; ++m; }
    }
  }
  sV[mat][role] = 0.0f;                       // first row phase sees v = 0
  asm volatile("s_wait_asynccnt 0x0" ::: "memory");
  __syncthreads();

  #pragma unroll 1
  for (int it = 0; it < N_ITERS; ++it) {
    // ---------- row phase: thread = row `role`, reduce over j ----------
    {
      const float* xr = &sx[mat * MAT_STRIDE + role * ROW_STRIDE];
      float a[36];
      float m0 = NEG, m1 = NEG, m2 = NEG, m3 = NEG;
      #pragma unroll
      for (int j = 0; j < 36; j += 4) {
        float t0 = __builtin_fmaf(K, xr[j + 0], -sV[mat][j + 0]); a[j + 0] = t0; m0 = fmaxf(m0, t0);
        float t1 = __builtin_fmaf(K, xr[j + 1], -sV[mat][j + 1]); a[j + 1] = t1; m1 = fmaxf(m1, t1);
        float t2 = __builtin_fmaf(K, xr[j + 2], -sV[mat][j + 2]); a[j + 2] = t2; m2 = fmaxf(m2, t2);
        float t3 = __builtin_fmaf(K, xr[j + 3], -sV[mat][j + 3]); a[j + 3] = t3; m3 = fmaxf(m3, t3);
      }
      float mx = fmaxf(fmaxf(m0, m1), fmaxf(m2, m3));
      float s0 = 0.f, s1 = 0.f, s2 = 0.f, s3 = 0.f;
      #pragma unroll
      for (int j = 0; j < 36; j += 4) {
        s0 += __builtin_amdgcn_exp2f(a[j + 0] - mx);
        s1 += __builtin_amdgcn_exp2f(a[j + 1] - mx);
        s2 += __builtin_amdgcn_exp2f(a[j + 2] - mx);
        s3 += __builtin_amdgcn_exp2f(a[j + 3] - mx);
      }
      sU[mat][role] = mx + __builtin_amdgcn_logf((s0 + s1) + (s2 + s3));
    }
    __syncthreads();

    // ---------- col phase: thread = column `role`, reduce over i ----------
    {
      const float* xc = &sx[mat * MAT_STRIDE + role];
      float a[36];
      float m0 = NEG, m1 = NEG, m2 = NEG, m3 = NEG;
      #pragma unroll
      for (int i = 0; i < 36; i += 4) {
        float t0 = __builtin_fmaf(K, xc[(i + 0) * ROW_STRIDE], -sU[mat][i + 0]); a[i + 0] = t0; m0 = fmaxf(m0, t0);
        float t1 = __builtin_fmaf(K, xc[(i + 1) * ROW_STRIDE], -sU[mat][i + 1]); a[i + 1] = t1; m1 = fmaxf(m1, t1);
        float t2 = __builtin_fmaf(K, xc[(i + 2) * ROW_STRIDE], -sU[mat][i + 2]); a[i + 2] = t2; m2 = fmaxf(m2, t2);
        float t3 = __builtin_fmaf(K, xc[(i + 3) * ROW_STRIDE], -sU[mat][i + 3]); a[i + 3] = t3; m3 = fmaxf(m3, t3);
      }
      float mx = fmaxf(fmaxf(m0, m1), fmaxf(m2, m3));
      float s0 = 0.f, s1 = 0.f, s2 = 0.f, s3 = 0.f;
      #pragma unroll
      for (int i = 0; i < 36; i += 4) {
        s0 += __builtin_amdgcn_exp2f(a[i + 0] - mx);
        s1 += __builtin_amdgcn_exp2f(a[i + 1] - mx);
        s2 += __builtin_amdgcn_exp2f(a[i + 2] - mx);
        s3 += __builtin_amdgcn_exp2f(a[i + 3] - mx);
      }
      sV[mat][role] = mx + __builtin_amdgcn_logf((s0 + s1) + (s2 + s3));
    }
    __syncthreads();
  }

  // ---------- output: 2^(K*x - U_i - V_j), coalesced NT stores ----------
  {
    float* gout = out + blockBase;
    int m = 0, i = mat, j = role;
    int f = tid;
    #pragma unroll
    for (int k = 0; k < 36; ++k) {
      float x = sx[m * MAT_STRIDE + i * ROW_STRIDE + j];
      float r = __builtin_amdgcn_exp2f(__builtin_fmaf(K, x, -(sU[m][i] + sV[m][j])));
      __builtin_nontemporal_store(r, &gout[f]);
      f += NTHREADS;
      i += 8; if (i >= 36) { i -= 36; ++m; }
    }
  }
}

extern "C" void kernel_launch(void* const* d_in, const int* in_sizes, int n_in,
                              void* d_out, int out_size, void* d_ws, size_t ws_size,
                              hipStream_t stream) {
  const float* in = (const float*)d_in[0];
  float* out = (float*)d_out;
  const int total  = in_sizes[0];                 // 65536 * 1296
  const int nmats  = total / ELEMS_PER_MAT;       // 65536
  const int blocks = nmats / MATS_PER_BLOCK;      // 8192
  hipLaunchKernelGGL(sinkhorn36_kernel, dim3(blocks), dim3(NTHREADS), 0, stream,
                     in, out);
}